// PointNetFeaturePropagation_23639499997597
// MI455X (gfx1250) — compile-verified
//
#include <hip/hip_runtime.h>
#include <hip/hip_bf16.h>

#define B_    8
#define N1_   2048
#define N2_   8192
#define C1_   128
#define C2_   256
#define INCH_ 384
#define MLP0_ 256
#define MLP1_ 128
#define EPS_  1e-5f

typedef __attribute__((ext_vector_type(2))) float v2f;
typedef __attribute__((ext_vector_type(8))) float v8f;

static __device__ __forceinline__ v8f wmma4(v2f a, v2f b, v8f c) {
  // D = A(16x4 f32) * B(4x16 f32) + C(16x16 f32)
  return __builtin_amdgcn_wmma_f32_16x16x4_f32(false, a, false, b, (short)0, c, false, false);
}

// ---------------------------------------------------------------------------
// K1: 1-NN argmin via WMMA. score = -2*x1.x2 + |x2|^2 (argmin-equivalent to d2).
// One wave per 16-row tile of xyz1; loops over N2 in 16-col tiles.
// ---------------------------------------------------------------------------
__global__ __launch_bounds__(32) void nn_argmin_kernel(
    const float* __restrict__ xyz1, const float* __restrict__ xyz2,
    int* __restrict__ idx) {
  const int tile = blockIdx.x;                 // B_ * (N1_/16) tiles
  const int b    = tile / (N1_ / 16);
  const int m0   = (tile % (N1_ / 16)) * 16;
  const int lane = threadIdx.x & 31;
  const int l    = lane & 15;
  const bool hi  = lane >= 16;

  // A tile (16x4): row M = m0+l; lanes<16 hold K=0,1; lanes>=16 hold K=2,3.
  // A = [x, y, z, 1]
  const float* p1 = xyz1 + ((size_t)b * N1_ + (m0 + l)) * 3;
  v2f A;
  A.x = hi ? p1[2] : p1[0];
  A.y = hi ? 1.0f  : p1[1];

  float minv[8];
  int   mini[8];
#pragma unroll
  for (int v = 0; v < 8; ++v) { minv[v] = 3.4e38f; mini[v] = 0; }

  const float* x2b = xyz2 + (size_t)b * N2_ * 3;
#pragma unroll 2
  for (int nt = 0; nt < N2_ / 16; ++nt) {
    const int n = nt * 16 + l;
    const float* p2 = x2b + (size_t)n * 3;
    const float x = p2[0], y = p2[1], z = p2[2];
    // B tile (4x16): col N = n; lanes<16 hold K=0,1; lanes>=16 hold K=2,3.
    // B = [-2x; -2y; -2z; |p2|^2]
    v2f Bv;
    Bv.x = hi ? (-2.0f * z)             : (-2.0f * x);
    Bv.y = hi ? (x * x + y * y + z * z) : (-2.0f * y);
    v8f c = {};
    c = wmma4(A, Bv, c);
    // C layout: c[v] -> M = m0 + v (+8 if hi), N = n (this lane's column)
#pragma unroll
    for (int v = 0; v < 8; ++v) {
      const float d = c[v];
      const bool take = d < minv[v];
      minv[v] = take ? d : minv[v];
      mini[v] = take ? n : mini[v];
    }
  }
  // argmin across the 16 lanes of each half (each half covers all 16 N columns);
  // branchless selects -> v_cndmask, no exec-mask branching.
#pragma unroll
  for (int off = 1; off < 16; off <<= 1) {
#pragma unroll
    for (int v = 0; v < 8; ++v) {
      const float ov = __shfl_xor(minv[v], off, 32);
      const int   oi = __shfl_xor(mini[v], off, 32);
      const bool take = (ov < minv[v]) | ((ov == minv[v]) & (oi < mini[v]));
      minv[v] = take ? ov : minv[v];
      mini[v] = take ? oi : mini[v];
    }
  }
  if (l == 0) {
#pragma unroll
    for (int v = 0; v < 8; ++v) {
      const int m = m0 + v + (hi ? 8 : 0);
      idx[(size_t)b * N1_ + m] = mini[v];
    }
  }
}

// ---------------------------------------------------------------------------
// K2: gather interpolated features + concat points1 -> NP (B, 384, N1)
// ---------------------------------------------------------------------------
__global__ void gather_concat_kernel(const float* __restrict__ points1,
                                     const float* __restrict__ points2,
                                     const int* __restrict__ idx,
                                     float* __restrict__ NP) {
  const int t = blockIdx.x * blockDim.x + threadIdx.x;
  const int total = B_ * INCH_ * N1_;
  if (t >= total) return;
  const int n = t % N1_;
  const int c = (t / N1_) % INCH_;
  const int b = t / (N1_ * INCH_);
  float v;
  if (c < C2_) {
    const int j = idx[b * N1_ + n];
    v = points2[((size_t)b * C2_ + c) * N2_ + j];
  } else {
    v = points1[((size_t)b * C1_ + (c - C2_)) * N1_ + n];
  }
  NP[(size_t)t] = v;
}

// ---------------------------------------------------------------------------
// K3/K5: conv1x1 as per-batch WMMA f32 GEMM.
// One wave computes a 16(M) x 64(N) strip: 4 accumulators reuse one A fragment
// per K-step (4 back-to-back independent WMMAs), K unrolled 2x for ILP.
// BN=true fuses BatchNorm+ReLU into the B-operand load (for the 2nd conv).
// ---------------------------------------------------------------------------
template <int MDIM, int KDIM, bool BN>
__global__ __launch_bounds__(256) void gemm_kernel(
    const float* __restrict__ W,     // (MDIM, KDIM) row-major
    const float* __restrict__ bias,  // (MDIM)
    const float* __restrict__ X,     // (B, KDIM, N1)
    const float* __restrict__ scale, // (KDIM) or unused
    const float* __restrict__ shift, // (KDIM) or unused
    float* __restrict__ Y) {         // (B, MDIM, N1)
  const int wave = threadIdx.x >> 5;
  const int lane = threadIdx.x & 31;
  const int tid  = blockIdx.x * 8 + wave;
  const int ntiles = N1_ / 64;       // 64-wide N strips
  const int mtiles = MDIM / 16;
  const int b  = tid / (mtiles * ntiles);
  const int r  = tid % (mtiles * ntiles);
  const int m0 = (r / ntiles) * 16;
  const int n0 = (r % ntiles) * 64;
  const int l  = lane & 15;
  const bool hi = lane >= 16;

  const float* Xb = X + (size_t)b * KDIM * N1_;
  v8f acc[4] = {};
#pragma unroll 2
  for (int k = 0; k < KDIM; k += 4) {
    const int ka = hi ? (k + 2) : k;
    // A fragment: row m0+l, K=ka..ka+1 -> single b64 load, reused by 4 WMMAs
    const v2f A = *(const v2f*)(W + (size_t)(m0 + l) * KDIM + ka);
    const float* xr0 = Xb + (size_t)ka * N1_ + n0 + l;
    const float* xr1 = Xb + (size_t)(ka + 1) * N1_ + n0 + l;
    float sc0 = 0.f, sh0 = 0.f, sc1 = 0.f, sh1 = 0.f;
    if (BN) { sc0 = scale[ka]; sh0 = shift[ka]; sc1 = scale[ka + 1]; sh1 = shift[ka + 1]; }
#pragma unroll
    for (int j = 0; j < 4; ++j) {
      float x0 = xr0[j * 16];
      float x1 = xr1[j * 16];
      if (BN) {
        x0 = fmaxf(0.0f, x0 * sc0 + sh0);
        x1 = fmaxf(0.0f, x1 * sc1 + sh1);
      }
      v2f Bv; Bv.x = x0; Bv.y = x1;
      acc[j] = wmma4(A, Bv, acc[j]);
    }
  }
#pragma unroll
  for (int j = 0; j < 4; ++j) {
#pragma unroll
    for (int v = 0; v < 8; ++v) {
      const int m = m0 + v + (hi ? 8 : 0);
      Y[((size_t)b * MDIM + m) * N1_ + (n0 + j * 16 + l)] = acc[j][v] + bias[m];
    }
  }
}

// ---------------------------------------------------------------------------
// K4/K6: per-channel BN stats over (B, N); fold into scale/shift.
// One block per channel; biased variance (matches jnp.var), EPS=1e-5.
// ---------------------------------------------------------------------------
template <int MDIM>
__global__ __launch_bounds__(256) void bn_stats_kernel(
    const float* __restrict__ Y,     // (B, MDIM, N1)
    const float* __restrict__ gamma, const float* __restrict__ beta,
    float* __restrict__ scale, float* __restrict__ shift) {
  const int c = blockIdx.x;
  __shared__ float ssum[256], ssq[256];
  float s = 0.0f, q = 0.0f;
  for (int i = threadIdx.x; i < B_ * N1_; i += 256) {
    const int b = i >> 11;           // N1_ == 2048
    const int n = i & (N1_ - 1);
    const float v = Y[((size_t)b * MDIM + c) * N1_ + n];
    s += v; q += v * v;
  }
  ssum[threadIdx.x] = s; ssq[threadIdx.x] = q;
  __syncthreads();
  for (int off = 128; off > 0; off >>= 1) {
    if (threadIdx.x < off) {
      ssum[threadIdx.x] += ssum[threadIdx.x + off];
      ssq[threadIdx.x]  += ssq[threadIdx.x + off];
    }
    __syncthreads();
  }
  if (threadIdx.x == 0) {
    const float inv  = 1.0f / (float)(B_ * N1_);
    const float mean = ssum[0] * inv;
    const float var  = ssq[0] * inv - mean * mean;
    const float sc   = gamma[c] * rsqrtf(var + EPS_);
    scale[c] = sc;
    shift[c] = beta[c] - mean * sc;
  }
}

// ---------------------------------------------------------------------------
// K7: final BN + ReLU -> d_out
// ---------------------------------------------------------------------------
__global__ void bn_apply_kernel(const float* __restrict__ Y,  // (B, MLP1, N1)
                                const float* __restrict__ scale,
                                const float* __restrict__ shift,
                                float* __restrict__ out) {
  const int t = blockIdx.x * blockDim.x + threadIdx.x;
  const int total = B_ * MLP1_ * N1_;
  if (t >= total) return;
  const int c = (t / N1_) % MLP1_;
  out[t] = fmaxf(0.0f, Y[t] * scale[c] + shift[c]);
}

// ---------------------------------------------------------------------------
extern "C" void kernel_launch(void* const* d_in, const int* in_sizes, int n_in,
                              void* d_out, int out_size, void* d_ws, size_t ws_size,
                              hipStream_t stream) {
  (void)in_sizes; (void)n_in; (void)out_size; (void)ws_size;
  const float* xyz1    = (const float*)d_in[0];
  const float* xyz2    = (const float*)d_in[1];
  const float* points1 = (const float*)d_in[2];
  const float* points2 = (const float*)d_in[3];
  const float* w0      = (const float*)d_in[4];
  const float* b0      = (const float*)d_in[5];
  const float* g0      = (const float*)d_in[6];
  const float* beta0   = (const float*)d_in[7];
  const float* w1      = (const float*)d_in[8];
  const float* b1      = (const float*)d_in[9];
  const float* g1      = (const float*)d_in[10];
  const float* beta1   = (const float*)d_in[11];

  // Workspace layout
  constexpr size_t off_idx = 0;
  constexpr size_t off_NP  = 65536;                                      // idx: 64 KB
  constexpr size_t szNP    = (size_t)B_ * INCH_ * N1_ * sizeof(float);   // 25.2 MB
  constexpr size_t off_Y0  = off_NP + szNP;
  constexpr size_t szY0    = (size_t)B_ * MLP0_ * N1_ * sizeof(float);   // 16.8 MB
  constexpr size_t off_Y1  = off_Y0 + szY0;
  constexpr size_t szY1    = (size_t)B_ * MLP1_ * N1_ * sizeof(float);   // 8.4 MB
  constexpr size_t off_s   = off_Y1 + szY1;

  char* ws = (char*)d_ws;
  int*   idx    = (int*)  (ws + off_idx);
  float* NP     = (float*)(ws + off_NP);
  float* Y0     = (float*)(ws + off_Y0);
  float* Y1     = (float*)(ws + off_Y1);
  float* scale0 = (float*)(ws + off_s);
  float* shift0 = scale0 + MLP0_;
  float* scale1 = shift0 + MLP0_;
  float* shift1 = scale1 + MLP1_;

  // K1: 1-NN argmin (WMMA distance tiles)
  nn_argmin_kernel<<<B_ * (N1_ / 16), 32, 0, stream>>>(xyz1, xyz2, idx);

  // K2: gather + concat
  {
    const int total = B_ * INCH_ * N1_;
    gather_concat_kernel<<<(total + 255) / 256, 256, 0, stream>>>(points1, points2, idx, NP);
  }

  // K3: conv0 (WMMA GEMM, 16x64 strips) + bias
  gemm_kernel<MLP0_, INCH_, false>
      <<<(B_ * (MLP0_ / 16) * (N1_ / 64)) / 8, 256, 0, stream>>>(
          w0, b0, NP, nullptr, nullptr, Y0);

  // K4: BN0 stats
  bn_stats_kernel<MLP0_><<<MLP0_, 256, 0, stream>>>(Y0, g0, beta0, scale0, shift0);

  // K5: conv1 (WMMA GEMM) with BN0+ReLU fused into B-load, + bias
  gemm_kernel<MLP1_, MLP0_, true>
      <<<(B_ * (MLP1_ / 16) * (N1_ / 64)) / 8, 256, 0, stream>>>(
          w1, b1, Y0, scale0, shift0, Y1);

  // K6: BN1 stats
  bn_stats_kernel<MLP1_><<<MLP1_, 256, 0, stream>>>(Y1, g1, beta1, scale1, shift1);

  // K7: final BN+ReLU -> output (B, 128, N1)
  {
    const int total = B_ * MLP1_ * N1_;
    bn_apply_kernel<<<(total + 255) / 256, 256, 0, stream>>>(Y1, scale1, shift1, (float*)d_out);
  }
}